// TimeDomainCBCWaveformGenerator_1425929142828
// MI455X (gfx1250) — compile-verified
//
#include <hip/hip_runtime.h>
#include <hip/hip_bf16.h>

// ---------------------------------------------------------------------------
// TaylorF2-style frequency-domain waveform generator for gfx1250 (MI455X).
//
// Output layout (floats, interleaved complex): [ hc (B*F*2) | hp (B*F*2) ].
// F and df are derived from out_size (chirplen is a power of two in the
// reference, so num_freqs = out_size/(4*B) and df = 1024/(F-1)) — no
// device->host sync, graph-capture safe.
//
// Roofline: ~268 MB of stores vs ~6 KB of loads -> HBM-store bound
// (~11.5 us floor at 23.3 TB/s). Hot loop: 1x v_log + 2x v_exp + sincos per
// element, coalesced nontemporal float2 stores (write-once stream, keep out
// of L2). Per-row scalars staged via the CDNA5 Tensor Data Mover
// (tensor_load_to_lds + s_wait_tensorcnt). No WMMA: workload has zero matrix
// structure, and the kernel is pinned to the store-bandwidth roofline.
// ---------------------------------------------------------------------------

#define D_MSUN 1.9884099021470415e30
#define D_G    6.67430e-11
#define D_C    299792458.0
#define D_PI   3.14159265358979323846

typedef unsigned int uint32_tt;
typedef uint32_tt v4u __attribute__((ext_vector_type(4)));
typedef int       v8i __attribute__((ext_vector_type(8)));
typedef int       v4i __attribute__((ext_vector_type(4)));
typedef float     v2f __attribute__((ext_vector_type(2)));

// --------------------------- per-row precompute ----------------------------
// ws layout per row (8 floats): acoef, pcoef, pfac, cosi, k0, k1, fstart, pad
__global__ __launch_bounds__(256)
void tf2_precompute(const float* __restrict__ mass1,
                    const float* __restrict__ mass2,
                    const float* __restrict__ s1z,
                    const float* __restrict__ s2z,
                    const float* __restrict__ dist,
                    const float* __restrict__ incl,
                    float df, int B, float* __restrict__ out)
{
    int i = blockIdx.x * blockDim.x + threadIdx.x;
    if (i >= B) return;

    double m1 = (double)mass1[i] * D_MSUN;
    double m2 = (double)mass2[i] * D_MSUN;
    double s1 = (double)s1z[i];
    double s2 = (double)s2z[i];
    double M  = m1 + m2;
    double nu = m1 * m2 / (M * M);
    double c3pow = D_C * D_C * D_C;

    // f_isco and f_min
    double f_isco = c3pow / (D_G * M * D_PI * 14.696938456699069); // 6^1.5
    double f_min  = fmin(f_isco, 20.0);

    // chirp-time bound at f_min
    double chi = fmax(fabs(s1), fabs(s2));
    double c0  = 5.0 * M * D_G / (256.0 * nu * c3pow);
    double c2  = 743.0 / 252.0 + 11.0 * nu / 3.0;
    double c3c = (226.0 / 15.0) * chi - 32.0 * D_PI / 5.0;
    double c4  = 3058673.0 / 508032.0 + 5429.0 * nu / 504.0 + 617.0 * nu * nu / 72.0;
    double v   = cbrt(D_PI * D_G * M * f_min) / D_C;
    double v2  = v * v;
    double tchirp = c0 / (v2 * v2 * v2 * v2) * (1.0 + (c2 + (c3c + c4 * v) * v) * v2);

    // chirp start frequency bound at (1+0.1)*tchirp
    double fstart = pow(c0 / (1.1 * tchirp), 0.375) * c3pow / (D_PI * D_G * M);

    // main-kernel constants
    double mc    = pow(nu, 0.6) * M;                             // chirp mass
    double acoef = pow(D_G * mc / c3pow, 5.0 / 6.0) * (D_C / (double)dist[i]);
    double pcoef = (3.0 / (128.0 * nu)) * pow(D_PI * D_G * M / c3pow, -5.0 / 3.0);

    float cosi = cosf(incl[i]);
    float pfac = 0.5f * (1.0f + cosi * cosi);
    float k0   = rintf((float)(fstart / (double)df));
    float k1   = rintf((float)(f_min  / (double)df));

    float* o = out + (size_t)i * 8;
    o[0] = (float)acoef;
    o[1] = (float)pcoef;
    o[2] = pfac;
    o[3] = cosi;
    o[4] = k0;
    o[5] = k1;
    o[6] = (float)fstart;
    o[7] = 0.0f;
}

// ------------------------------ main kernel --------------------------------
// grid: (ceil(F/2048), B), 256 threads, 8 bins/thread (stride 256).
__global__ __launch_bounds__(256)
void tf2_main(const float* __restrict__ params, float* __restrict__ out,
              int F, float df, float theta, int B)
{
    __shared__ float sp[8];
    const int b = blockIdx.y;

    if (threadIdx.x == 0) {
        // Stage this row's 8 params into LDS with the Tensor Data Mover.
        // D# per CDNA5 ISA: count=1, type=2,
        // data_size=4B, 1-D tile/tensor of 8 elements, stride 8.
        uint32_tt lds_addr = (uint32_tt)(unsigned long long)(void*)sp;
        unsigned long long ga =
            (unsigned long long)(const void*)(params + (size_t)b * 8);

        v4u g0;
        g0[0] = 1u;                                         // count=1, user mode
        g0[1] = lds_addr;                                   // lds_addr
        g0[2] = (uint32_tt)ga;                              // global_addr[31:0]
        g0[3] = (uint32_tt)((ga >> 32) & 0x1FFFFFFull) | (2u << 30); // [56:32]+type=2

        v8i g1;
        g1[0] = 0x00020000;        // workgroup_mask=0, data_size=2 (4 bytes)
        g1[1] = (int)(8u << 16);   // tensor_dim0 = 8 (low 16 bits at [63:48])
        g1[2] = 0;                 // tensor_dim0 hi, tensor_dim1 lo = 0
        g1[3] = (int)(8u << 16);   // tile_dim0 = 8
        g1[4] = 0;                 // tile_dim1 = tile_dim2 = 0 (unused -> 1D)
        g1[5] = 8;                 // tensor_dim0_stride = 8
        g1[6] = 0;
        g1[7] = 0;

        v4i z4 = {0, 0, 0, 0};
        v8i z8 = {0, 0, 0, 0, 0, 0, 0, 0};
        __builtin_amdgcn_tensor_load_to_lds(g0, g1, z4, z4, z8, 0);
        __builtin_amdgcn_s_wait_tensorcnt(0);
    }
    __syncthreads();

    const float acoef  = sp[0];
    const float pcoef  = sp[1];
    const float pfac   = sp[2];
    const float cfac   = sp[3];
    const float k0f    = sp[4];
    const float k1f    = sp[5];
    const float fstart = sp[6];

    float* __restrict__ hc = out + (size_t)b * (size_t)F * 2;
    float* __restrict__ hp = hc + (size_t)B * (size_t)F * 2;

    const float invspan = 1.0f / fmaxf(k1f - k0f, 1.0f);
    const int   kbase   = blockIdx.x * 2048 + threadIdx.x;
    const int   Fm1     = F - 1;

    auto emit = [&](int k) {
        float kf = (float)k;
        float f  = kf * df;
        float fg = fmaxf(f, df);                 // guard k = 0

        // f^{-5/3} and f^{-7/6} via one v_log_f32 + two v_exp_f32
        float t     = __log2f(fg);
        float phase = pcoef * __builtin_exp2f(-1.66666666666666667f * t);
        float amp   = acoef * __builtin_exp2f(-1.16666666666666667f * t);

        // combined angle: coalescence phase shift minus SPA phase
        float alpha = fmaf(theta, kf, -phase);
        float sa, ca;
        __sincosf(alpha, &sa, &ca);

        // cosine taper over [k0, k1]
        float scale = 1.0f;
        if (kf >= k0f && kf <= k1f) {
            float w = 3.14159265358979323846f * (kf - k0f) * invspan;
            scale = 0.5f - 0.5f * __cosf(w);
        }
        // zero below per-row fstart; zero the Nyquist bin
        if (f < fstart || k == Fm1) scale = 0.0f;

        float g  = amp * scale;
        float ap = g * pfac;
        float ac = g * cfac;

        // h*e^{i*theta*k} = amp*(cos a + i sin a);  hc = -i*cosi*h_shift
        v2f vc; vc.x = ac * sa; vc.y = -ac * ca;
        v2f vp; vp.x = ap * ca; vp.y =  ap * sa;

        // write-once stream: nontemporal, 8B stores are always aligned and
        // lane-contiguous -> fully coalesced 256B/wave
        __builtin_nontemporal_store(vc, (v2f*)(hc + 2 * (size_t)k));
        __builtin_nontemporal_store(vp, (v2f*)(hp + 2 * (size_t)k));
    };

    if (kbase + 7 * 256 < F) {
        // fast path: whole block strictly inside the row, no bounds checks
#pragma unroll
        for (int j = 0; j < 8; ++j)
            emit(kbase + j * 256);
    } else {
        // tail block (only the last blockIdx.x per row)
#pragma unroll
        for (int j = 0; j < 8; ++j) {
            int k = kbase + j * 256;
            if (k < F) emit(k);
        }
    }
}

// ------------------------------- launcher ----------------------------------
extern "C" void kernel_launch(void* const* d_in, const int* in_sizes, int n_in,
                              void* d_out, int out_size, void* d_ws, size_t ws_size,
                              hipStream_t stream)
{
    const float* mass1 = (const float*)d_in[0];
    const float* mass2 = (const float*)d_in[1];
    const float* s1z   = (const float*)d_in[2];
    const float* s2z   = (const float*)d_in[3];
    const float* dist  = (const float*)d_in[4];
    const float* incl  = (const float*)d_in[5];

    const int B = in_sizes[0];
    // out_size counts float elements: 2 outputs * B * F * 2 (re,im)
    const int F = out_size / (4 * B);
    if (F < 2) return;

    const float df    = 1024.0f / (float)(F - 1);   // exact: F-1 is a power of 2
    const float theta = (float)D_PI * df;           // 2*pi*df*tshift, tshift = 0.5

    float* ws_params = (float*)d_ws;                // 8 floats per row

    tf2_precompute<<<(B + 255) / 256, 256, 0, stream>>>(
        mass1, mass2, s1z, s2z, dist, incl, df, B, ws_params);

    dim3 grid((unsigned)((F + 2047) / 2048), (unsigned)B);
    tf2_main<<<grid, 256, 0, stream>>>(ws_params, (float*)d_out, F, df, theta, B);
}